// KVCache_72275709657687
// MI455X (gfx1250) — compile-verified
//
#include <hip/hip_runtime.h>
#include <hip/hip_bf16.h>
#include <stdint.h>

// ---------------------------------------------------------------------------
// Problem constants (from reference): B=2,H=32,S=4096,D=128,U=16
// ---------------------------------------------------------------------------
constexpr int B = 2, H = 32, S = 4096, D = 128, U = 16;
constexpr int GROUP = 128, RESID = 128;
constexpr float LEVELS = 15.0f;
constexpr int Tq = S - RESID - (S % GROUP);   // 3968
constexpr int NG = Tq / GROUP;                // 31 groups per (b,h)
constexpr int BH = B * H;                     // 64

// flat f32 offsets of each output in d_out (reference return order)
constexpr long long N_QK   = (long long)BH * (Tq / 4) * D;  // packed k codes
constexpr long long N_KSC  = (long long)BH * NG * D;        // k scale / min
constexpr long long N_QV   = (long long)BH * (D / 4) * Tq;  // packed v codes
constexpr long long N_VSC  = (long long)BH * Tq;            // v scale / min
constexpr long long N_RES  = (long long)BH * (2 * RESID + 1) * D;
constexpr long long N_FULL = (long long)BH * S * D;

constexpr long long O_QK  = 0;
constexpr long long O_KSC = O_QK  + N_QK;
constexpr long long O_KMN = O_KSC + N_KSC;
constexpr long long O_QV  = O_KMN + N_KSC;
constexpr long long O_VSC = O_QV  + N_QV;
constexpr long long O_VMN = O_VSC + N_VSC;
constexpr long long O_KR  = O_VMN + N_VSC;
constexpr long long O_VR  = O_KR  + N_RES;
constexpr long long O_KF  = O_VR  + N_RES;
constexpr long long O_VF  = O_KF  + N_FULL;

typedef float        f4    __attribute__((ext_vector_type(4)));
typedef unsigned int u32x4 __attribute__((ext_vector_type(4)));
typedef unsigned int u32x8 __attribute__((ext_vector_type(8)));

// ---------------------------------------------------------------------------
// CDNA5 async global->LDS copy (ASYNCcnt-tracked). LDS operand is the 32-bit
// LDS offset == low 32 bits of the generic pointer (aperture rule, ISA 10.2).
// ---------------------------------------------------------------------------
__device__ __forceinline__ void async_ld_b128(const float* gsrc, const float* lds_dst) {
    unsigned l = (unsigned)(unsigned long long)(const void*)lds_dst;
    asm volatile("global_load_async_to_lds_b128 %0, %1, off"
                 :
                 : "v"(l), "v"(gsrc)
                 : "memory");
}

__device__ __forceinline__ void wait_async0() {
    asm volatile("s_wait_asynccnt 0" ::: "memory");
}

// ---------------------------------------------------------------------------
// CDNA5 Tensor Data Mover: 1D contiguous tile load, TENSORcnt-tracked.
// D# group0 (4 SGPRs): count=1 | lds_addr | global_addr[56:0] | type=2.
// D# group1 (8 SGPRs): data_size=4B, tensor_dim0=tile_dim0=n elements.
// ---------------------------------------------------------------------------
__device__ __forceinline__ void tdm_load_1d(const float* gsrc, const float* lds_dst,
                                            unsigned nelem) {
    const unsigned lds = (unsigned)(unsigned long long)(const void*)lds_dst;
    const unsigned long long ga = (unsigned long long)gsrc;

    union { unsigned a[4]; u32x4 v; } g0;
    g0.a[0] = 1u;                                           // count=1, user mode
    g0.a[1] = lds;                                          // lds_addr (bytes)
    g0.a[2] = (unsigned)ga;                                 // global_addr[31:0]
    g0.a[3] = (unsigned)((ga >> 32) & 0x01FFFFFFu)          // global_addr[56:32]
              | (2u << 30);                                 // type=2 ("image")

    union { unsigned a[8]; u32x8 v; } g1;
    g1.a[0] = 2u << 16;                                     // data_size=2 (4 bytes)
    g1.a[1] = (nelem & 0xFFFFu) << 16;                      // tensor_dim0[15:0]
    g1.a[2] = (nelem >> 16) | (1u << 16);                   // tensor_dim0[31:16], tensor_dim1=1
    g1.a[3] = (nelem & 0xFFFFu) << 16;                      // tile_dim0
    g1.a[4] = 0;                                            // tile_dim1=0, tile_dim2=0
    g1.a[5] = nelem;                                        // tensor_dim0_stride[31:0]
    g1.a[6] = 0;                                            // stride hi, dim1_stride lo
    g1.a[7] = 0;

    asm volatile("tensor_load_to_lds %0, %1" :: "s"(g0.v), "s"(g1.v) : "memory");
}

__device__ __forceinline__ float quant_code(float v, float mn, float inv) {
    return fminf(fmaxf(rintf((v - mn) * inv), 0.0f), LEVELS);
}

// ---------------------------------------------------------------------------
// K quantization: one block per (b,h,seq-group). Tile (128 seq x 128 d) is
// contiguous in memory -> single TDM descriptor load into LDS (issued by
// wave 0 only; TDM ignores EXEC, so gate with a scalar branch via
// readfirstlane). Lane d then reduces its column and packs 4 codes/u16.
// ---------------------------------------------------------------------------
__global__ __launch_bounds__(128) void kquant_kernel(const float* __restrict__ k,
                                                     float* __restrict__ out) {
    __shared__ __align__(16) float sk[GROUP * D];   // 64 KB
    const int tid = threadIdx.x;
    const int bh  = blockIdx.x / NG;
    const int gi  = blockIdx.x % NG;
    const float* src = k + ((long long)bh * S + (long long)gi * GROUP) * D;

    // wave 0 issues one TDM DMA for the whole 64 KB tile
    if (__builtin_amdgcn_readfirstlane(tid >> 5) == 0) {
        tdm_load_1d(src, sk, GROUP * D);
        __builtin_amdgcn_s_wait_tensorcnt(0);
    }
    __syncthreads();

    const int d = tid;
    float mn = sk[d], mx = sk[d];
#pragma unroll 4
    for (int s = 1; s < GROUP; ++s) {
        const float v = sk[s * D + d];
        mn = fminf(mn, v);
        mx = fmaxf(mx, v);
    }
    const float scale = fmaxf((mx - mn) * (1.0f / LEVELS), 1e-6f);
    const float inv   = 1.0f / scale;
    const long long sc_idx = ((long long)bh * NG + gi) * D + d;
    out[O_KSC + sc_idx] = scale;
    out[O_KMN + sc_idx] = mn;

    float* qk = out + O_QK + ((long long)bh * (Tq / 4) + (long long)gi * (GROUP / 4)) * D;
#pragma unroll
    for (int sp = 0; sp < GROUP / 4; ++sp) {
        unsigned pk = 0;
#pragma unroll
        for (int j = 0; j < 4; ++j) {
            const float q = quant_code(sk[(sp * 4 + j) * D + d], mn, inv);
            pk |= ((unsigned)q) << (4 * j);
        }
        __builtin_nontemporal_store((float)pk, qk + (long long)sp * D + d);
    }
}

// ---------------------------------------------------------------------------
// V quantization: one block per (b,h,seq-tile). Async-stage into padded LDS
// (row stride 132 floats) so lane t reads its row without systematic bank
// conflicts; reduce over D, write packed codes transposed (coalesced on Tq).
// ---------------------------------------------------------------------------
constexpr int VSTR = D + 4;   // 132 floats, keeps 16B alignment per row

__global__ __launch_bounds__(128) void vquant_kernel(const float* __restrict__ v,
                                                     float* __restrict__ out) {
    __shared__ __align__(16) float sv[GROUP * VSTR];   // ~66 KB
    const int tid = threadIdx.x;
    const int bh  = blockIdx.x / NG;
    const int ti  = blockIdx.x % NG;
    const int t0  = ti * GROUP;
    const float* src = v + ((long long)bh * S + t0) * D;

#pragma unroll
    for (int i = 0; i < 32; ++i) {
        const int c4 = i * 128 + tid;       // float4 chunk within tile
        const int s  = c4 >> 5;             // 32 chunks per row
        const int c  = c4 & 31;
        async_ld_b128(src + c4 * 4, &sv[s * VSTR + c * 4]);
    }
    wait_async0();
    __syncthreads();

    const int t = tid;                       // seq position within tile
    const float* row = &sv[t * VSTR];
    float mn = row[0], mx = row[0];
#pragma unroll 4
    for (int dd = 1; dd < D; ++dd) {
        const float x = row[dd];
        mn = fminf(mn, x);
        mx = fmaxf(mx, x);
    }
    const float scale = fmaxf((mx - mn) * (1.0f / LEVELS), 1e-6f);
    const float inv   = 1.0f / scale;
    out[O_VSC + (long long)bh * Tq + t0 + t] = scale;
    out[O_VMN + (long long)bh * Tq + t0 + t] = mn;

#pragma unroll
    for (int dp = 0; dp < D / 4; ++dp) {
        unsigned pk = 0;
#pragma unroll
        for (int j = 0; j < 4; ++j) {
            const float q = quant_code(row[dp * 4 + j], mn, inv);
            pk |= ((unsigned)q) << (4 * j);
        }
        __builtin_nontemporal_store(
            (float)pk, out + O_QV + ((long long)bh * (D / 4) + dp) * Tq + t0 + t);
    }
}

// ---------------------------------------------------------------------------
// Residual tails: rows [0,128) copy k/v tail, rows [128,257) zero.
// 16-byte vector loads / non-temporal stores.
// ---------------------------------------------------------------------------
__global__ void resid_kernel(const float* __restrict__ k, const float* __restrict__ v,
                             float* __restrict__ out) {
    const long long n4 = N_RES / 4;
    const long long i  = (long long)blockIdx.x * blockDim.x + threadIdx.x;
    if (i >= n4) return;
    const int c   = (int)(i % (D / 4));         // float4 column within row
    long long r   = i / (D / 4);
    const int row = (int)(r % (2 * RESID + 1));
    const int bh  = (int)(r / (2 * RESID + 1));
    f4 kv = {0.0f, 0.0f, 0.0f, 0.0f};
    f4 vv = {0.0f, 0.0f, 0.0f, 0.0f};
    if (row < RESID) {
        const long long src4 = (((long long)bh * S + Tq + row) * D) / 4 + c;
        kv = ((const f4*)k)[src4];
        vv = ((const f4*)v)[src4];
    }
    __builtin_nontemporal_store(kv, (f4*)(out + O_KR) + i);
    __builtin_nontemporal_store(vv, (f4*)(out + O_VR) + i);
}

// ---------------------------------------------------------------------------
// Zero-fill k_full/v_full (256 MB streaming) with non-temporal 16B stores.
// ---------------------------------------------------------------------------
__global__ void zero_full_kernel(float* __restrict__ out) {
    f4* p = (f4*)(out + O_KF);             // 16B-aligned (O_KF % 4 == 0)
    const long long n4 = (2 * N_FULL) / 4;
    long long i = (long long)blockIdx.x * blockDim.x + threadIdx.x;
    const long long stride = (long long)gridDim.x * blockDim.x;
    const f4 z = {0.0f, 0.0f, 0.0f, 0.0f};
    for (; i < n4; i += stride) __builtin_nontemporal_store(z, p + i);
}

// ---------------------------------------------------------------------------
// Scatter k_new/v_new at per-batch residual offsets into k_full/v_full.
// ---------------------------------------------------------------------------
__global__ void scatter_kernel(const float* __restrict__ kn, const float* __restrict__ vn,
                               const int* __restrict__ cs, const int* __restrict__ qcs,
                               float* __restrict__ out) {
    const long long n = (long long)B * H * U * D;
    const long long i = (long long)blockIdx.x * blockDim.x + threadIdx.x;
    if (i >= n) return;
    const int d = (int)(i % D);
    long long r = i / D;
    const int u = (int)(r % U); r /= U;
    const int h = (int)(r % H);
    const int b = (int)(r / H);
    const int off = cs[b] - qcs[b];
    const long long dst = (((long long)(b * H + h)) * S + off + u) * D + d;
    out[O_KF + dst] = kn[i];
    out[O_VF + dst] = vn[i];
}

// ---------------------------------------------------------------------------
extern "C" void kernel_launch(void* const* d_in, const int* in_sizes, int n_in,
                              void* d_out, int out_size, void* d_ws, size_t ws_size,
                              hipStream_t stream) {
    const float* k   = (const float*)d_in[0];
    const float* v   = (const float*)d_in[1];
    const float* kn  = (const float*)d_in[2];
    const float* vn  = (const float*)d_in[3];
    const int*   cs  = (const int*)d_in[4];
    const int*   qcs = (const int*)d_in[5];
    float* out = (float*)d_out;

    // Streaming zero-fill first; scatter afterwards (same stream => ordered).
    zero_full_kernel<<<4096, 256, 0, stream>>>(out);

    kquant_kernel<<<BH * NG, 128, 0, stream>>>(k, out);
    vquant_kernel<<<BH * NG, 128, 0, stream>>>(v, out);

    const long long n_res4 = N_RES / 4;
    resid_kernel<<<(int)((n_res4 + 255) / 256), 256, 0, stream>>>(k, v, out);

    const long long n_new = (long long)B * H * U * D;
    scatter_kernel<<<(int)((n_new + 255) / 256), 256, 0, stream>>>(kn, vn, cs, qcs, out);
}